// GCNNet_53815940219570
// MI455X (gfx1250) — compile-verified
//
#include <hip/hip_runtime.h>
#include <hip/hip_bf16.h>

typedef float v2f __attribute__((ext_vector_type(2)));
typedef float v8f __attribute__((ext_vector_type(8)));

#define BN_EPS 1e-5f

// ---------------------------------------------------------------------------
// Graph-normalization precompute
// ---------------------------------------------------------------------------
__global__ void k_fill1(float* __restrict__ p, int n) {
    int i = blockIdx.x * blockDim.x + threadIdx.x;
    if (i < n) p[i] = 1.0f;  // self-loop contributes 1 to every degree
}

__global__ void k_deg(const int* __restrict__ col, float* __restrict__ deg, int E) {
    int e = blockIdx.x * blockDim.x + threadIdx.x;
    if (e < E) atomicAdd(&deg[col[e]], 1.0f);
}

__global__ void k_dinv(const float* __restrict__ deg, float* __restrict__ dinv, int n) {
    int i = blockIdx.x * blockDim.x + threadIdx.x;
    if (i < n) dinv[i] = rsqrtf(deg[i]);  // deg >= 1 always (self-loops)
}

__global__ void k_norm(const int* __restrict__ row, const int* __restrict__ col,
                       const float* __restrict__ dinv, float* __restrict__ nm, int E) {
    int e = blockIdx.x * blockDim.x + threadIdx.x;
    if (e < E) nm[e] = dinv[row[e]] * dinv[col[e]];
}

// ---------------------------------------------------------------------------
// fp32 GEMM via V_WMMA_F32_16X16X4_F32.
// One wave (32 threads) computes a 16(M) x 64(N) output tile: the A fragment
// is loaded once per K-step and reused across 4 N-subtiles.
// VGPR layouts per CDNA5 ISA 7.12.2:
//   A 16x4 f32:  v0 = {K=0 | K=2}, v1 = {K=1 | K=3}  (lanes 0-15 | 16-31)
//   B 4x16 f32:  v0 = {row 0 | row 2}, v1 = {row 1 | row 3}
//   C 16x16 f32: vj = {M=j, N=lane | M=j+8, N=lane-16}
// ---------------------------------------------------------------------------
__global__ __launch_bounds__(32) void k_gemm_wmma(
    const float* __restrict__ X, const int* __restrict__ gidx,
    const float* __restrict__ W, float* __restrict__ Y,
    int N, int din, int dout) {
    const int lane = threadIdx.x;
    const int half = lane >> 4;   // 0: lanes 0-15, 1: lanes 16-31
    const int l    = lane & 15;
    const int m0   = blockIdx.x * 16;
    const int n0   = blockIdx.y * 64;

    int arow = m0 + l;
    bool valid = arow < N;
    if (!valid) arow = N - 1;
    if (gidx) arow = gidx[arow];
    const float* __restrict__ Arow = X + (size_t)arow * din;

    v8f c0 = {}, c1 = {}, c2 = {}, c3 = {};
    for (int k = 0; k < din; k += 4) {
        const int kk = k + half * 2;  // this lane holds K = kk, kk+1
        v2f a;
        a.x = Arow[kk + 0];
        a.y = Arow[kk + 1];
        const float* __restrict__ W0 = W + (size_t)(kk + 0) * dout + n0 + l;
        const float* __restrict__ W1 = W + (size_t)(kk + 1) * dout + n0 + l;
        v2f b0, b1, b2, b3;
        b0.x = W0[0];  b0.y = W1[0];
        b1.x = W0[16]; b1.y = W1[16];
        b2.x = W0[32]; b2.y = W1[32];
        b3.x = W0[48]; b3.y = W1[48];
        c0 = __builtin_amdgcn_wmma_f32_16x16x4_f32(false, a, false, b0, (short)0, c0, false, false);
        c1 = __builtin_amdgcn_wmma_f32_16x16x4_f32(false, a, false, b1, (short)0, c1, false, false);
        c2 = __builtin_amdgcn_wmma_f32_16x16x4_f32(false, a, false, b2, (short)0, c2, false, false);
        c3 = __builtin_amdgcn_wmma_f32_16x16x4_f32(false, a, false, b3, (short)0, c3, false, false);
    }

    // Store: VGPR j of C holds (M = m0+j+half*8, N = n0 + 16*sub + l)
    #pragma unroll
    for (int j = 0; j < 8; ++j) {
        int m = m0 + j + half * 8;
        if (m < N) {
            float* __restrict__ Yr = Y + (size_t)m * dout + n0 + l;
            Yr[0]  = c0[j];
            Yr[16] = c1[j];
            Yr[32] = c2[j];
            Yr[48] = c3[j];
        }
    }
}

// ---------------------------------------------------------------------------
// Aggregation: out = xl * dinv^2 (self-loop) + bias, then edge atomics on top.
// ---------------------------------------------------------------------------
__global__ void k_selfinit(const float* __restrict__ xl, const float* __restrict__ dinv,
                           const float* __restrict__ bias, float* __restrict__ out,
                           int total, int dshift) {
    int i = blockIdx.x * blockDim.x + threadIdx.x;
    if (i < total) {
        int node = i >> dshift;
        int f    = i & ((1 << dshift) - 1);
        float dv = dinv[node];
        out[i] = xl[i] * (dv * dv) + bias[f];
    }
}

// 1 thread per (edge, feature). All lanes of a given edge read a contiguous
// 4*dout-byte span of xl (coalesced) and issue one global_atomic_add_f32.
// Working set fits in the 192MB L2, so these resolve at L2.
__global__ void k_edge(const int* __restrict__ row, const int* __restrict__ col,
                       const float* __restrict__ nm, const float* __restrict__ xl,
                       float* __restrict__ out, int E, int dshift) {
    long long t = (long long)blockIdx.x * blockDim.x + threadIdx.x;
    int e = (int)(t >> dshift);
    int f = (int)(t & ((1 << dshift) - 1));
    if (e < E) {
        int r = row[e], c = col[e];
        float v = xl[((size_t)r << dshift) + f] * nm[e];
        atomicAdd(&out[((size_t)c << dshift) + f], v);
    }
}

// ---------------------------------------------------------------------------
// BatchNorm (training stats, population variance) + ReLU
// ---------------------------------------------------------------------------
__global__ __launch_bounds__(256) void k_bnstats(const float* __restrict__ x,
                                                 float* __restrict__ sums,
                                                 float* __restrict__ sumsq, int N) {
    int f = threadIdx.x;                 // feature 0..255 (H == 256)
    int r0 = blockIdx.x * 256;
    int rend = min(r0 + 256, N);
    float s = 0.f, s2 = 0.f;
    for (int r = r0; r < rend; ++r) {
        float v = x[(size_t)r * 256 + f];
        s += v;
        s2 += v * v;
    }
    atomicAdd(&sums[f], s);
    atomicAdd(&sumsq[f], s2);
}

__global__ void k_bnfinal(const float* __restrict__ sums, const float* __restrict__ sumsq,
                          const float* __restrict__ g, const float* __restrict__ be,
                          float* __restrict__ scale, float* __restrict__ shift, int N) {
    int f = threadIdx.x;
    float inv_n = 1.0f / (float)N;
    float m = sums[f] * inv_n;
    float v = sumsq[f] * inv_n - m * m;
    float sc = g[f] * rsqrtf(v + BN_EPS);
    scale[f] = sc;
    shift[f] = be[f] - m * sc;
}

__global__ void k_bnrelu(const float* __restrict__ x, const float* __restrict__ scale,
                         const float* __restrict__ shift, float* __restrict__ y, int total) {
    int i = blockIdx.x * blockDim.x + threadIdx.x;
    if (i < total) {
        int f = i & 255;
        float v = x[i] * scale[f] + shift[f];
        y[i] = v > 0.f ? v : 0.f;
    }
}

// ---------------------------------------------------------------------------
// Host-side layer driver
// ---------------------------------------------------------------------------
static void run_conv(const float* x, const int* gidx, const float* W, const float* bias,
                     float* xl, float* outbuf, const int* row, const int* col,
                     const float* nm, const float* dinv,
                     int N, int E, int din, int dout, int dshift, hipStream_t stream) {
    dim3 gg((N + 15) / 16, dout / 64);
    k_gemm_wmma<<<gg, 32, 0, stream>>>(x, gidx, W, xl, N, din, dout);
    int total = N * dout;
    k_selfinit<<<(total + 255) / 256, 256, 0, stream>>>(xl, dinv, bias, outbuf, total, dshift);
    long long et = (long long)E << dshift;
    k_edge<<<(unsigned)((et + 255) / 256), 256, 0, stream>>>(row, col, nm, xl, outbuf, E, dshift);
}

extern "C" void kernel_launch(void* const* d_in, const int* in_sizes, int n_in,
                              void* d_out, int out_size, void* d_ws, size_t ws_size,
                              hipStream_t stream) {
    const int*   x_indices = (const int*)d_in[0];
    const int*   ei        = (const int*)d_in[1];   // (2, E) row-major
    const float* emb       = (const float*)d_in[2];
    const float* W1 = (const float*)d_in[3];  const float* b1 = (const float*)d_in[4];
    const float* g1 = (const float*)d_in[5];  const float* be1 = (const float*)d_in[6];
    const float* W2 = (const float*)d_in[7];  const float* b2 = (const float*)d_in[8];
    const float* g2 = (const float*)d_in[9];  const float* be2 = (const float*)d_in[10];
    const float* W3 = (const float*)d_in[11]; const float* b3 = (const float*)d_in[12];

    const int N = in_sizes[0];
    const int E = in_sizes[1] / 2;
    const int D = in_sizes[2] / N;        // 128
    const int H = in_sizes[4];            // 256
    const int O = in_sizes[12];           // 64

    const int* row = ei;          // source
    const int* col = ei + E;      // target (aggregation index)

    // Workspace carve-up (floats)
    float* ws    = (float*)d_ws;
    float* deg   = ws;                         // N
    float* dinv  = deg   + N;                  // N
    float* nm    = dinv  + N;                  // E
    float* xl    = nm    + E;                  // N*H
    float* agg   = xl    + (size_t)N * H;      // N*H
    float* xact  = agg   + (size_t)N * H;      // N*H
    float* sums  = xact  + (size_t)N * H;      // H
    float* sumsq = sums  + H;                  // H
    float* scale = sumsq + H;                  // H
    float* shift = scale + H;                  // H

    float* out = (float*)d_out;

    // --- normalization: deg (with self-loops), dinv, per-edge norm ---
    k_fill1<<<(N + 255) / 256, 256, 0, stream>>>(deg, N);
    k_deg<<<(E + 255) / 256, 256, 0, stream>>>(col, deg, E);
    k_dinv<<<(N + 255) / 256, 256, 0, stream>>>(deg, dinv, N);
    k_norm<<<(E + 255) / 256, 256, 0, stream>>>(row, col, dinv, nm, E);

    // --- layer 1: conv(emb[x_indices], W1, b1) -> BN -> ReLU ---
    run_conv(emb, x_indices, W1, b1, xl, agg, row, col, nm, dinv, N, E, D, H, /*dshift=*/8, stream);
    hipMemsetAsync(sums, 0, 2 * H * sizeof(float), stream);
    k_bnstats<<<(N + 255) / 256, 256, 0, stream>>>(agg, sums, sumsq, N);
    k_bnfinal<<<1, H, 0, stream>>>(sums, sumsq, g1, be1, scale, shift, N);
    k_bnrelu<<<(N * H + 255) / 256, 256, 0, stream>>>(agg, scale, shift, xact, N * H);

    // --- layer 2: conv(x, W2, b2) -> BN -> ReLU ---
    run_conv(xact, nullptr, W2, b2, xl, agg, row, col, nm, dinv, N, E, H, H, /*dshift=*/8, stream);
    hipMemsetAsync(sums, 0, 2 * H * sizeof(float), stream);
    k_bnstats<<<(N + 255) / 256, 256, 0, stream>>>(agg, sums, sumsq, N);
    k_bnfinal<<<1, H, 0, stream>>>(sums, sumsq, g2, be2, scale, shift, N);
    k_bnrelu<<<(N * H + 255) / 256, 256, 0, stream>>>(agg, scale, shift, xact, N * H);

    // --- layer 3: conv(x, W3, b3) straight into d_out ---
    run_conv(xact, nullptr, W3, b3, xl, out, row, col, nm, dinv, N, E, H, O, /*dshift=*/6, stream);
}